// InteractionBlock_21182778704569
// MI455X (gfx1250) — compile-verified
//
#include <hip/hip_runtime.h>
#include <hip/hip_bf16.h>
#include <math.h>

// Problem constants from the reference
#define B_  8
#define V_  256
#define H_  128
#define M_TOTAL (B_ * V_)        // 2048 rows
#define NT_TILES (H_ / 16)       // 8 column tiles

typedef float v2f __attribute__((ext_vector_type(2)));
typedef float v4f __attribute__((ext_vector_type(4)));
typedef float v8f __attribute__((ext_vector_type(8)));

__device__ __forceinline__ float silu_f(float x) {
    return x / (1.0f + __expf(-x));
}

// ---------------------------------------------------------------------------
// Kernel 1: h0 = silu(h @ W_pre + b_pre)  -> cat[:, 0:128]
//           s  = silu(h @ W_skip + b_skip) -> outS (d_out, consumed later)
// One wave per 16x16 output tile, shared A-fragment for both GEMMs.
// Grid: 1024 tiles / 4 waves = 256 blocks of 128 threads.
// ---------------------------------------------------------------------------
__global__ __launch_bounds__(128)
void gemm_pre_skip(const float* __restrict__ h,
                   const float* __restrict__ Wpre,  const float* __restrict__ bpre,
                   const float* __restrict__ Wskip, const float* __restrict__ bskip,
                   float* __restrict__ cat, float* __restrict__ outS) {
    const int lane = threadIdx.x & 31;
    const int wave = threadIdx.x >> 5;
    const int tile = blockIdx.x * 4 + wave;
    const int mt = tile >> 3;          // 0..127  (row tile)
    const int nt = tile & 7;           // 0..7    (col tile)
    const int l    = lane & 15;
    const int half = lane >> 4;        // 0 or 1 -> K sub-pair / row-half of C

    const int row = mt * 16 + l;       // A row for this lane
    const int col = nt * 16 + l;       // B/C column for this lane

    v8f acc0 = {};   // pre
    v8f acc1 = {};   // skip

    const float* arow = h + (size_t)row * H_;
    #pragma unroll 4
    for (int k = 0; k < H_; k += 4) {
        const int kk = k + half * 2;
        v2f a;  a.x  = arow[kk];                 a.y  = arow[kk + 1];
        v2f b0; b0.x = Wpre [(size_t)kk * H_ + col];  b0.y = Wpre [(size_t)(kk + 1) * H_ + col];
        v2f b1; b1.x = Wskip[(size_t)kk * H_ + col];  b1.y = Wskip[(size_t)(kk + 1) * H_ + col];
        acc0 = __builtin_amdgcn_wmma_f32_16x16x4_f32(false, a, false, b0, (short)0, acc0, false, false);
        acc1 = __builtin_amdgcn_wmma_f32_16x16x4_f32(false, a, false, b1, (short)0, acc1, false, false);
    }

    const float bias0 = bpre[col];
    const float bias1 = bskip[col];
    const int rowbase = mt * 16 + half * 8;      // C/D: VGPR r -> row rowbase+r
    #pragma unroll
    for (int r = 0; r < 8; ++r) {
        const float v0 = silu_f(acc0[r] + bias0);
        const float v1 = silu_f(acc1[r] + bias1);
        cat [(size_t)(rowbase + r) * (2 * H_) + nt * 16 + l] = v0;  // h0 -> left half
        outS[(size_t)(rowbase + r) * H_        + nt * 16 + l] = v1; // skip branch
    }
}

// ---------------------------------------------------------------------------
// Kernel 2: neigh[b,i,h] = max_j ( graph[b,i,j] ? 0 : e[b,i,j,h] * h0[b,j,h] )
// One block per (b,i); 256 threads = 32 lanes x 4 channels x 8 j-partitions.
// Each lane owns 4 contiguous channels -> global_load_b128 (512 B per wave
// request), e streamed non-temporally (268 MB read-once, don't pollute L2;
// h0 is 1 MB and stays L2-resident). 8-way partial-max folded through LDS.
// Writes neigh into cat[:, 128:256].
// ---------------------------------------------------------------------------
__global__ __launch_bounds__(256)
void aggregate_max(const float* __restrict__ e,
                   const int*   __restrict__ graph,
                   float* __restrict__ cat) {
    const int bi = blockIdx.x;          // b*V + i
    const int b  = bi >> 8;             // V_ = 256

    __shared__ int   gmask[V_];
    __shared__ float red[8 * H_];       // 8 partitions x 128 channels

    const int t = threadIdx.x;
    gmask[t] = graph[(size_t)bi * V_ + t];
    __syncthreads();

    const int c0   = (t & 31) * 4;      // first of 4 channels for this lane
    const int part = t >> 5;            // 0..7 j-partition

    const float* erow = e + (size_t)bi * V_ * H_;
    const float* h0b  = cat + (size_t)b * V_ * (2 * H_);   // h0[b, j, :], stride 2H

    v4f acc = { -INFINITY, -INFINITY, -INFINITY, -INFINITY };
    const int j0 = part * (V_ / 8);     // 32 j's per partition
    #pragma unroll 4
    for (int j = j0; j < j0 + V_ / 8; ++j) {
        const v4f ev = __builtin_nontemporal_load(
            (const v4f*)(erow + (size_t)j * H_ + c0));
        const v4f hv = *(const v4f*)(h0b + (size_t)j * (2 * H_) + c0);
        const bool masked = (gmask[j] != 0);
        const float m0 = masked ? 0.0f : ev.x * hv.x;
        const float m1 = masked ? 0.0f : ev.y * hv.y;
        const float m2 = masked ? 0.0f : ev.z * hv.z;
        const float m3 = masked ? 0.0f : ev.w * hv.w;
        acc.x = fmaxf(acc.x, m0);
        acc.y = fmaxf(acc.y, m1);
        acc.z = fmaxf(acc.z, m2);
        acc.w = fmaxf(acc.w, m3);
    }
    *(v4f*)(red + part * H_ + c0) = acc;
    __syncthreads();

    if (t < H_) {
        float m = red[t];
        #pragma unroll
        for (int p = 1; p < 8; ++p)
            m = fmaxf(m, red[p * H_ + t]);
        cat[(size_t)bi * (2 * H_) + H_ + t] = m;           // neigh -> right half
    }
}

// ---------------------------------------------------------------------------
// Kernel 3: h1 = silu(cat @ W_post + b_post); out = silu(s + h1)
// s is currently in d_out; read it, fuse the final silu, overwrite d_out.
// ---------------------------------------------------------------------------
__global__ __launch_bounds__(128)
void gemm_post(const float* __restrict__ cat,
               const float* __restrict__ Wpost, const float* __restrict__ bpost,
               float* __restrict__ out) {
    const int lane = threadIdx.x & 31;
    const int wave = threadIdx.x >> 5;
    const int tile = blockIdx.x * 4 + wave;
    const int mt = tile >> 3;
    const int nt = tile & 7;
    const int l    = lane & 15;
    const int half = lane >> 4;

    const int row = mt * 16 + l;
    const int col = nt * 16 + l;

    v8f acc = {};
    const float* arow = cat + (size_t)row * (2 * H_);
    #pragma unroll 4
    for (int k = 0; k < 2 * H_; k += 4) {
        const int kk = k + half * 2;
        v2f a;  a.x = arow[kk];                       a.y = arow[kk + 1];
        v2f bb; bb.x = Wpost[(size_t)kk * H_ + col];  bb.y = Wpost[(size_t)(kk + 1) * H_ + col];
        acc = __builtin_amdgcn_wmma_f32_16x16x4_f32(false, a, false, bb, (short)0, acc, false, false);
    }

    const float bias = bpost[col];
    const int rowbase = mt * 16 + half * 8;
    #pragma unroll
    for (int r = 0; r < 8; ++r) {
        const size_t idx = (size_t)(rowbase + r) * H_ + nt * 16 + l;
        const float h1 = silu_f(acc[r] + bias);
        const float s  = out[idx];
        out[idx] = silu_f(s + h1);
    }
}

// ---------------------------------------------------------------------------
extern "C" void kernel_launch(void* const* d_in, const int* in_sizes, int n_in,
                              void* d_out, int out_size, void* d_ws, size_t ws_size,
                              hipStream_t stream) {
    const float* h      = (const float*)d_in[0];
    const float* e      = (const float*)d_in[1];
    const int*   graph  = (const int*)  d_in[2];
    const float* W_pre  = (const float*)d_in[3];
    const float* b_pre  = (const float*)d_in[4];
    const float* W_post = (const float*)d_in[5];
    const float* b_post = (const float*)d_in[6];
    const float* W_skip = (const float*)d_in[7];
    const float* b_skip = (const float*)d_in[8];

    float* out = (float*)d_out;                   // 2048 x 128
    float* cat = (float*)d_ws;                    // 2048 x 256 (h0 | neigh), 2 MB

    const int tiles  = (M_TOTAL / 16) * NT_TILES; // 1024
    const int blocks = tiles / 4;                 // 256 blocks, 4 waves each

    gemm_pre_skip<<<blocks, 128, 0, stream>>>(h, W_pre, b_pre, W_skip, b_skip, cat, out);
    aggregate_max<<<M_TOTAL, 256, 0, stream>>>(e, graph, cat);
    gemm_post<<<blocks, 128, 0, stream>>>(cat, W_post, b_post, out);
}